// BatchPSCN_88699664597435
// MI455X (gfx1250) — compile-verified
//
#include <hip/hip_runtime.h>
#include <hip/hip_bf16.h>
#include <cstdint>

typedef __attribute__((ext_vector_type(2))) float v2f;
typedef __attribute__((ext_vector_type(8))) float v8f;

// Exact pointee type the async-LDS builtin expects (per hipcc diagnostic):
// 'int __attribute__((__vector_size__(4*sizeof(int))))'
typedef int v4i __attribute__((vector_size(16)));
typedef __attribute__((address_space(1))) v4i gv4i;   // prints as '__device__'
typedef __attribute__((address_space(3))) v4i lv4i;   // prints as '__shared__'

#define BS   128
#define NNOD 4000
#define DIM  128
#define NB   10
#define SEQ  400
#define C1   32
#define C2   32
#define TILES_PER_B (SEQ / 16)   // 25
#define NTILE (BS * TILES_PER_B) // 3200
#define Y2_FLOATS (BS * SEQ * C2)          // 1,638,400 floats (6.55 MB)
#define W1P_FLOATS (NB * C1 * DIM)         // 40,960 floats

#if __has_builtin(__builtin_amdgcn_global_load_async_to_lds_b128)
#define HAVE_ASYNC_LDS 1
#else
#define HAVE_ASYNC_LDS 0
#endif

#if __has_builtin(__builtin_amdgcn_s_wait_asynccnt)
#define WAIT_ASYNC(n) __builtin_amdgcn_s_wait_asynccnt(n)
#else
#define WAIT_ASYNC(n) asm volatile("s_wait_asynccnt %0" :: "i"(n) : "memory")
#endif

// Pre-pack conv1_w [O=32][C=128][K=10] -> w1p [kk][o][c] (channel contiguous),
// so each WMMA B-fragment (two consecutive channels) is a single b64 load.
__global__ __launch_bounds__(256)
void pscn_pack_w1(const float* __restrict__ w1, float* __restrict__ w1p) {
    int tid = blockIdx.x * 256 + threadIdx.x;   // enumerate dst [kk][o][c]
    if (tid >= W1P_FLOATS) return;
    int kk = tid / (C1 * DIM);
    int r  = tid % (C1 * DIM);
    int o  = r / DIM;
    int c  = r % DIM;
    w1p[tid] = w1[o * (DIM * NB) + c * NB + kk];
}

// Fused: gather + conv1 (as GEMM, v_wmma_f32_16x16x4_f32) + bias/ReLU
//        + conv2 (1x1 GEMM, WMMA) + bias/ReLU -> ws y2[m*32 + c2]
// One wave32 per 16-row tile (16 consecutive seq positions of one batch).
// Gather rows stream into a double-buffered LDS tile with async global->LDS
// loads (ASYNCcnt), overlapping the next tap's gather with WMMA compute.
__global__ __launch_bounds__(32)
void pscn_conv_kernel(const int*   __restrict__ recep,      // [BS, NNOD]
                      const float* __restrict__ node_feat,  // [BS*NNOD, DIM]
                      const float* __restrict__ w1p,        // [NB][C1][DIM] packed
                      const float* __restrict__ b1,         // [C1]
                      const float* __restrict__ w2,         // [C2, C1] (1x1)
                      const float* __restrict__ b2,         // [C2]
                      float*       __restrict__ y2)         // [BS*SEQ, C2]
{
    __shared__ int   sidx[16 * NB];            // [kk][row] gather row ids
    __shared__ float sA[2][16 * DIM];          // double-buffered tap tile (2x8KB)
    __shared__ float sY[16 * C1];              // conv1 output tile (2KB)

    const int lane = threadIdx.x;              // 0..31
    const int t    = blockIdx.x;
    const int b    = t / TILES_PER_B;
    const int s0   = (t % TILES_PER_B) * 16;

    // gather indices: 160 ints
    for (int e = lane; e < 16 * NB; e += 32) {
        int kk  = e >> 4;
        int row = e & 15;
        int pos = (s0 + row) * NB + kk;
        sidx[e] = recep[b * NNOD + pos] + b * NNOD;
    }
    __syncthreads();

    const int n = lane & 15;                   // N index within 16-wide tile
    const int g = lane >> 4;                   // half-wave group

    v8f acc0 = {};                             // C/D tile, columns o = 0..15
    v8f acc1 = {};                             // columns o = 16..31

#if HAVE_ASYNC_LDS
    // one async b128 per row: 32 lanes x 16B = 512B = one full feature row
    auto issue_tap = [&](int kk) {
        float* dstbuf = sA[kk & 1];
        #pragma unroll
        for (int i = 0; i < 16; ++i) {
            const float* src = node_feat + (size_t)sidx[kk * 16 + i] * DIM + lane * 4;
            float*       dst = dstbuf + i * DIM + lane * 4;
            __builtin_amdgcn_global_load_async_to_lds_b128(
                (gv4i*)(uintptr_t)src,
                (lv4i*)(uint32_t)(uintptr_t)dst, 0, 0);
        }
    };
    issue_tap(0);
#endif

    for (int kk = 0; kk < NB; ++kk) {
        const float* base = sA[kk & 1];
#if HAVE_ASYNC_LDS
        if (kk + 1 < NB) {
            // buffer (kk+1)&1 was last read by tap kk-1; ensure those DS reads
            // retired before the async engine overwrites it
            asm volatile("s_wait_dscnt 0x0" ::: "memory");
            issue_tap(kk + 1);
            WAIT_ASYNC(16);                    // in-order: oldest 16 (= tap kk) done
        } else {
            WAIT_ASYNC(0);
        }
        asm volatile("" ::: "memory");
#else
        // sync fallback: stage via VGPRs
        for (int j = lane; j < 16 * DIM / 4; j += 32) {
            int row = j >> 5;
            int c4  = j & 31;
            const float4* src =
                (const float4*)(node_feat + (size_t)sidx[kk * 16 + row] * DIM) + c4;
            ((float4*)sA[kk & 1])[row * 32 + c4] = *src;
        }
        __syncthreads();
#endif

        // per-lane packed-weight rows: channel-contiguous -> b64 fragment loads
        const v2f* wrow0 = (const v2f*)(w1p + (size_t)(kk * C1 + n)      * DIM);
        const v2f* wrow1 = (const v2f*)(w1p + (size_t)(kk * C1 + n + 16) * DIM);
        const v2f* arow  = (const v2f*)(base + n * DIM);

        // K = 128 per tap, 32 WMMA steps of K=4
        #pragma unroll 8
        for (int kq = 0; kq < DIM / 4; ++kq) {
            v2f a   = arow[kq * 2 + g];        // A: K pair {4kq+2g, 4kq+2g+1}
            v2f bf0 = wrow0[kq * 2 + g];       // B: same K pair, o = n
            v2f bf1 = wrow1[kq * 2 + g];       //                o = n + 16
            acc0 = __builtin_amdgcn_wmma_f32_16x16x4_f32(
                false, a, false, bf0, (short)0, acc0, false, false);
            acc1 = __builtin_amdgcn_wmma_f32_16x16x4_f32(
                false, a, false, bf1, (short)0, acc1, false, false);
        }
#if !HAVE_ASYNC_LDS
        __syncthreads();
#endif
    }

    // bias + ReLU; bounce y1 tile through LDS into A-fragment order
    {
        float bia0 = b1[n], bia1 = b1[n + 16];
        #pragma unroll
        for (int r = 0; r < 8; ++r) {
            int row = r + 8 * g;               // C/D layout: VGPR r, group g -> M=r+8g
            sY[row * C1 + n]      = fmaxf(acc0[r] + bia0, 0.0f);
            sY[row * C1 + n + 16] = fmaxf(acc1[r] + bia1, 0.0f);
        }
    }
    __syncthreads();

    // conv2 (1x1): [16 x 32] x [32 x 32], 8 WMMA steps; w2 is channel-contiguous
    v8f d0 = {}, d1 = {};
    {
        const v2f* arow  = (const v2f*)(sY + n * C1);
        const v2f* wrow0 = (const v2f*)(w2 + n * C1);
        const v2f* wrow1 = (const v2f*)(w2 + (n + 16) * C1);
        #pragma unroll
        for (int kq = 0; kq < C1 / 4; ++kq) {
            v2f a   = arow[kq * 2 + g];
            v2f bf0 = wrow0[kq * 2 + g];
            v2f bf1 = wrow1[kq * 2 + g];
            d0 = __builtin_amdgcn_wmma_f32_16x16x4_f32(
                false, a, false, bf0, (short)0, d0, false, false);
            d1 = __builtin_amdgcn_wmma_f32_16x16x4_f32(
                false, a, false, bf1, (short)0, d1, false, false);
        }
    }

    // bias + ReLU + store y2 tile: y2[(b*SEQ + s)*32 + o]
    {
        float bia0 = b2[n], bia1 = b2[n + 16];
        #pragma unroll
        for (int r = 0; r < 8; ++r) {
            int row = r + 8 * g;
            size_t m = (size_t)b * SEQ + s0 + row;
            y2[m * C2 + n]      = fmaxf(d0[r] + bia0, 0.0f);
            y2[m * C2 + n + 16] = fmaxf(d1[r] + bia1, 0.0f);
        }
    }
}

// Deterministic per-batch fc reduction: out[b] = fc_b + sum_{c,s} y2[b,c,s]*fc_w[c*SEQ+s]
__global__ __launch_bounds__(256)
void pscn_fc_kernel(const float* __restrict__ y2,    // [BS*SEQ, C2]
                    const float* __restrict__ fc_w,  // [1, C2*SEQ]
                    const float* __restrict__ fc_b,  // [1]
                    float*       __restrict__ out)   // [BS, 1]
{
    const int b   = blockIdx.x;
    const int tid = threadIdx.x;
    const float* rowp = y2 + (size_t)b * (SEQ * C2);
    float sum = 0.0f;
    for (int j = tid; j < SEQ * C2; j += 256) {
        int s = j >> 5;                        // y2 layout: s-major, 32 channels per s
        int c = j & 31;
        sum += rowp[j] * fc_w[c * SEQ + s];
    }
    __shared__ float red[256];
    red[tid] = sum;
    __syncthreads();
    for (int st = 128; st > 0; st >>= 1) {
        if (tid < st) red[tid] += red[tid + st];
        __syncthreads();
    }
    if (tid == 0) out[b] = red[0] + fc_b[0];
}

extern "C" void kernel_launch(void* const* d_in, const int* in_sizes, int n_in,
                              void* d_out, int out_size, void* d_ws, size_t ws_size,
                              hipStream_t stream) {
    (void)in_sizes; (void)n_in; (void)out_size; (void)ws_size;
    const int*   recep     = (const int*)  d_in[0];
    const float* node_feat = (const float*)d_in[1];
    const float* conv1_w   = (const float*)d_in[2];
    const float* conv1_b   = (const float*)d_in[3];
    const float* conv2_w   = (const float*)d_in[4];
    const float* conv2_b   = (const float*)d_in[5];
    const float* fc_w      = (const float*)d_in[6];
    const float* fc_b      = (const float*)d_in[7];
    float* out = (float*)d_out;
    float* y2  = (float*)d_ws;                       // 6.55 MB
    float* w1p = (float*)d_ws + Y2_FLOATS;           // +160 KB packed weights

    pscn_pack_w1<<<(W1P_FLOATS + 255) / 256, 256, 0, stream>>>(conv1_w, w1p);
    pscn_conv_kernel<<<NTILE, 32, 0, stream>>>(recep, node_feat, w1p, conv1_b,
                                               conv2_w, conv2_b, y2);
    pscn_fc_kernel<<<BS, 256, 0, stream>>>(y2, fc_w, fc_b, out);
}